// Binarize_49168785604695
// MI455X (gfx1250) — compile-verified
//
#include <hip/hip_runtime.h>
#include <stdint.h>

#define B_ 16
#define T_ 800000
#define W_ 800
#define WPR (T_ / 32)          // 25000 bitmask words per row
#define TILE_W 4096            // words per LDS tile (16 KB); even, so u64 view is exact
#define NTILES ((WPR + TILE_W - 1) / TILE_W)   // 7 (last tile: 424 words = 212 u64)
#define SPAN_CAP 4096          // true bound is ~2*T/W+1 ~= 2001

// ---------- CDNA5 async global->LDS copy (ASYNCcnt path) ----------
__device__ __forceinline__ void async_copy_b128(const uint32_t* g, uint32_t* lds_generic) {
#if defined(__gfx1250__)
  // LDS aperture: generic addr[31:0] is the LDS byte offset (ISA 10.2)
  uint32_t lofs = (uint32_t)(uintptr_t)lds_generic;
  asm volatile("global_load_async_to_lds_b128 %0, %1, off"
               :: "v"(lofs), "v"(g) : "memory");
#else
  ((uint4*)lds_generic)[0] = ((const uint4*)g)[0];
#endif
}
__device__ __forceinline__ void wait_async_le32() {
#if defined(__gfx1250__)
  asm volatile("s_wait_asynccnt 32" ::: "memory");
#endif
}
__device__ __forceinline__ void wait_async_0() {
#if defined(__gfx1250__)
  asm volatile("s_wait_asynccnt 0" ::: "memory");
#endif
}

// ---------- Kernel 1: threshold + ballot-pack; 128 samples per wave-iteration ----------
__global__ __launch_bounds__(256) void k_binarize(const float* __restrict__ x,
                                                  uint32_t* __restrict__ bits) {
  const int NCHUNK = (B_ * WPR) / 4;               // 100000 chunks of 128 samples
  const int wpb = 256 / 32;                        // waves per block
  const int wave = blockIdx.x * wpb + (threadIdx.x >> 5);
  const int lane = threadIdx.x & 31;
  const int total_waves = gridDim.x * wpb;

  for (int c = wave; c < NCHUNK; c += total_waves) {
    const long long base = (long long)c * 128;
    __builtin_prefetch(x + base + (long long)total_waves * 128 + lane * 4, 0, 3);
    uint32_t m0, m1, m2, m3;
    m0 = (uint32_t)__ballot(x[base +  0 + lane] > 0.5f ? 1 : 0);
    m1 = (uint32_t)__ballot(x[base + 32 + lane] > 0.5f ? 1 : 0);
    m2 = (uint32_t)__ballot(x[base + 64 + lane] > 0.5f ? 1 : 0);
    m3 = (uint32_t)__ballot(x[base + 96 + lane] > 0.5f ? 1 : 0);
    uint32_t w = (lane == 0) ? m0 : (lane == 1) ? m1 : (lane == 2) ? m2 : m3;
    if (lane < 4) bits[c * 4 + lane] = w;          // 16 B coalesced store
  }
}

// ---------- Kernel 2: one wave/row, async-LDS double-buffered 64-bit RLE scan ----------
__global__ __launch_bounds__(32) void k_scan(const uint32_t* __restrict__ bits,
                                             uint32_t* __restrict__ spans,
                                             uint32_t* __restrict__ counts) {
  __shared__ uint32_t tile[2][TILE_W];   // 32 KB double buffer
  const int row  = blockIdx.x;
  const int lane = threadIdx.x;
  const uint32_t* rowbits = bits + (size_t)row * WPR;
  uint32_t* rowspans = spans + (size_t)row * SPAN_CAP;

  // streaming scan state (lane 0 registers)
  int cur = 0, has_prev = 0, prev_val = 0, cnt = 0, last_val = -1;
  uint32_t rlen = 0, act = 0, nact = 0, group_start = 0;
  bool first = true;

  auto emit = [&](uint32_t endpos, uint32_t val) {
    if (cnt > 0 && last_val == (int)val) {
      rowspans[cnt - 1] = (endpos << 1) | val;               // merge same-value spans
    } else if (cnt < SPAN_CAP) {
      rowspans[cnt++] = (endpos << 1) | val;
      last_val = (int)val;
    } else {                                                  // unreachable; defensive
      rowspans[SPAN_CAP - 1] = ((uint32_t)T_ << 1) | (uint32_t)(last_val & 1);
    }
  };
  auto end_run = [&](uint32_t endpos) {
    if (rlen == 0) return;
    uint32_t startpos = endpos - rlen;
    if (rlen >= (uint32_t)W_) {                               // long run: flush group, keep run
      if (startpos > group_start) {
        uint32_t total = act + nact;
        uint32_t d = (total < (uint32_t)W_) ? (has_prev ? (uint32_t)prev_val : 0u)
                                            : (act > nact ? 1u : 0u);
        emit(startpos, d);
      }
      emit(endpos, (uint32_t)cur);
      has_prev = 1; prev_val = cur;
      act = 0; nact = 0; group_start = endpos;
    } else {                                                  // short run: accumulate group
      if (cur) act += rlen; else nact += rlen;
    }
  };

  // prologue: kick off tile 0 async copy (32 async ops per tile)
  for (int i = lane; i < TILE_W / 4; i += 32)
    async_copy_b128(rowbits + (size_t)i * 4, &tile[0][i * 4]);

  for (int ti = 0; ti < NTILES; ++ti) {
    if (ti + 1 < NTILES) {
      const uint32_t* gn = rowbits + (size_t)(ti + 1) * TILE_W; // may overrun row; stays in ws
      uint32_t* ln = tile[(ti + 1) & 1];
      for (int i = lane; i < TILE_W / 4; i += 32)
        async_copy_b128(gn + (size_t)i * 4, &ln[i * 4]);
      wait_async_le32();   // in-order completion: tile ti done, tile ti+1 may be in flight
    } else {
      wait_async_0();
    }
    asm volatile("" ::: "memory");

    if (lane == 0) {
      const uint64_t* lb = reinterpret_cast<const uint64_t*>(tile[ti & 1]);
      int wend = WPR - ti * TILE_W;
      if (wend > TILE_W) wend = TILE_W;
      wend >>= 1;                                             // u64 count (always exact)
      const uint32_t tile_base = (uint32_t)(ti * TILE_W) * 32u;
      for (int w = 0; w < wend; ++w) {
        uint64_t word = lb[w];
        uint32_t abs_base = tile_base + (uint32_t)w * 64u;
        if (first) { cur = (int)(word & 1ull); first = false; }
        int p = 0;
        while (p < 64) {
          uint64_t y = word >> p;
          if (cur) y = ~y;                                    // boundaries become set bits
          int n = 64 - p;
          if (n < 64) y &= (~0ull >> (64 - n));
          if (y == 0ull) { rlen += (uint32_t)n; p = 64; }     // run continues through word
          else {
            int t = (int)__builtin_ctzll(y);
            rlen += (uint32_t)t; p += t;
            end_run(abs_base + (uint32_t)p);                  // run boundary
            cur ^= 1; rlen = 0;
          }
        }
      }
    }
  }

  if (lane == 0) {
    end_run((uint32_t)T_);                                    // final run
    if ((uint32_t)T_ > group_start) {                         // trailing group
      uint32_t total = act + nact;
      uint32_t d = (total < (uint32_t)W_) ? (has_prev ? (uint32_t)prev_val : 0u)
                                          : (act > nact ? 1u : 0u);
      emit((uint32_t)T_, d);
    }
    if (cnt == 0) { rowspans[0] = ((uint32_t)T_ << 1); cnt = 1; }
    counts[row] = (uint32_t)cnt;
  }
}

// ---------- Kernel 3: expand spans -> float output (binary search in LDS) ----------
__device__ __forceinline__ int lower_span(const uint32_t* s, int n, uint32_t t) {
  int lo = 0, hi = n - 1;
  while (lo < hi) {
    int mid = (lo + hi) >> 1;
    if ((s[mid] >> 1) > t) hi = mid; else lo = mid + 1;
  }
  return lo;
}

__global__ __launch_bounds__(256) void k_expand(const uint32_t* __restrict__ spans,
                                                const uint32_t* __restrict__ counts,
                                                float* __restrict__ out) {
  __shared__ uint32_t sspan[SPAN_CAP];  // 16 KB
  const int row = blockIdx.y;
  int n = (int)counts[row];
  if (n < 1) n = 1;
  if (n > SPAN_CAP) n = SPAN_CAP;
  const uint32_t* rs = spans + (size_t)row * SPAN_CAP;
  for (int i = threadIdx.x; i < n; i += blockDim.x) sspan[i] = rs[i];
  __syncthreads();

  const int QPR = T_ / 4;   // 200000 float4 per row
  int q = blockIdx.x * blockDim.x + threadIdx.x;
  if (q >= QPR) return;
  uint32_t t0 = (uint32_t)q * 4u;
  int idx = lower_span(sspan, n, t0);
  float4 o;
  float* of = &o.x;
#pragma unroll
  for (int j = 0; j < 4; ++j) {
    uint32_t t = t0 + (uint32_t)j;
    while (idx < n - 1 && (sspan[idx] >> 1) <= t) ++idx;
    of[j] = (sspan[idx] & 1u) ? 1.0f : 0.0f;
  }
  reinterpret_cast<float4*>(out)[(size_t)row * QPR + q] = o;
}

extern "C" void kernel_launch(void* const* d_in, const int* in_sizes, int n_in,
                              void* d_out, int out_size, void* d_ws, size_t ws_size,
                              hipStream_t stream) {
  (void)in_sizes; (void)n_in; (void)out_size; (void)ws_size;
  const float* x = (const float*)d_in[0];
  float* out = (float*)d_out;
  uint32_t* ws = (uint32_t*)d_ws;
  // ws layout (words): [bits: B_*WPR][counts: 16][spans: B_*SPAN_CAP]  (~1.8 MB total)
  uint32_t* bits   = ws;
  uint32_t* counts = ws + (size_t)B_ * WPR;
  uint32_t* spans  = ws + (size_t)B_ * WPR + 16;

  k_binarize<<<2048, 256, 0, stream>>>(x, bits);
  k_scan<<<B_, 32, 0, stream>>>(bits, spans, counts);
  dim3 g3((T_ / 4 + 255) / 256, B_);
  k_expand<<<g3, 256, 0, stream>>>(spans, counts, out);
}